// GCN_34961033790072
// MI455X (gfx1250) — compile-verified
//
#include <hip/hip_runtime.h>
#include <hip/hip_bf16.h>

typedef float v2f __attribute__((ext_vector_type(2)));
typedef float v8f __attribute__((ext_vector_type(8)));

#define BN_EPS 1e-5f

// ---------------- small utility kernels ----------------

__global__ void fill_f32(float* __restrict__ p, long long n, float v) {
    long long gid = (long long)blockIdx.x * blockDim.x + threadIdx.x;
    if (gid < n) p[gid] = v;
}

__global__ void deg_count(const long long* __restrict__ dst, float* __restrict__ deg,
                          long long E) {
    long long gid = (long long)blockIdx.x * blockDim.x + threadIdx.x;
    if (gid < E) {
        long long d = dst[gid];
        atomicAdd(&deg[d], 1.0f);
    }
}

__global__ void make_dinv(const float* __restrict__ deg, float* __restrict__ dinv,
                          long long n) {
    long long gid = (long long)blockIdx.x * blockDim.x + threadIdx.x;
    if (gid < n) dinv[gid] = rsqrtf(deg[gid]);   // deg >= 1 (self-loop) always
}

// ---------------- WMMA f32 GEMM: H[N x COLS] = X[N x K] @ W[K x COLS] ----------------
// One wave per 16x16 output tile. N must be a multiple of 16 (100000 = 6250*16).
// COLS may be non-multiple-of-16 (40): masked via per-lane select, EXEC stays full
// around the WMMA itself.

template <int COLS>
__global__ __launch_bounds__(256) void gemm_wmma_f32(const float* __restrict__ X,
                                                     const float* __restrict__ W,
                                                     float* __restrict__ H,
                                                     int N, int K, int colTiles) {
    const int wave = threadIdx.x >> 5;
    const int lane = threadIdx.x & 31;
    const int tile = blockIdx.x * (blockDim.x >> 5) + wave;
    const int rowTile = tile / colTiles;
    const int colTile = tile - rowTile * colTiles;
    const int row0 = rowTile * 16;
    if (row0 >= N) return;                        // wave-uniform exit

    const int m  = lane & 15;                     // row within tile (A), col within tile (B)
    const int kq = (lane >> 4) << 1;              // 0 for lanes 0-15, 2 for lanes 16-31
    const int col = colTile * 16 + m;
    const bool colOk = (col < COLS);
    const int colC = colOk ? col : (COLS - 1);

    const float* __restrict__ xrow = X + (size_t)(row0 + m) * K;

    v8f acc = {};
    for (int k = 0; k < K; k += 4) {
        v2f a, b;
        // A 16x4 layout: lane m holds X[row0+m][k+kq], X[row0+m][k+kq+1]
        a.x = xrow[k + kq];
        a.y = xrow[k + kq + 1];
        // B 4x16 layout (mirror): lane m holds W[k+kq][col], W[k+kq+1][col]
        float b0 = W[(size_t)(k + kq) * COLS + colC];
        float b1 = W[(size_t)(k + kq + 1) * COLS + colC];
        b.x = colOk ? b0 : 0.0f;
        b.y = colOk ? b1 : 0.0f;
        acc = __builtin_amdgcn_wmma_f32_16x16x4_f32(
            /*neg_a=*/false, a, /*neg_b=*/false, b,
            /*c_mod=*/(short)0, acc, /*reuse_a=*/false, /*reuse_b=*/false);
    }

    // C/D layout: VGPR r -> M = r + 8*(lane>=16), N = lane&15
    if (colOk) {
        const int rbase = (lane >> 4) << 3;
#pragma unroll
        for (int r = 0; r < 8; ++r) {
            H[(size_t)(row0 + rbase + r) * COLS + col] = acc[r];
        }
    }
}

// ---------------- edge gather/scale/scatter-add ----------------
// One thread per (edge, channel): channel index is the fast dimension so the
// C consecutive lanes of a wave hit contiguous floats of h[s] / acc[d] (L2-resident).

template <int C>
__global__ void edge_scatter(const float* __restrict__ H,
                             const long long* __restrict__ src,
                             const long long* __restrict__ dst,
                             const float* __restrict__ dinv,
                             float* __restrict__ acc, long long total) {
    long long gid = (long long)blockIdx.x * blockDim.x + threadIdx.x;
    if (gid >= total) return;
    long long e = gid / C;
    int c = (int)(gid - e * (long long)C);
    long long s = src[e];
    long long d = dst[e];
    float nrm = dinv[s] * dinv[d];
    atomicAdd(&acc[d * C + c], H[s * C + c] * nrm);
}

// ---------------- self-loop + bias + BN(eval) + ReLU, fused ----------------
// out may alias acc (pure elementwise read->write per thread).

template <int C>
__global__ void node_finish(const float* __restrict__ H,
                            const float* __restrict__ acc,
                            const float* __restrict__ dinv,
                            const float* __restrict__ bias,
                            const float* __restrict__ gamma,
                            const float* __restrict__ beta,
                            const float* __restrict__ mean,
                            const float* __restrict__ var,
                            float* __restrict__ out, long long total) {
    long long gid = (long long)blockIdx.x * blockDim.x + threadIdx.x;
    if (gid >= total) return;
    long long i = gid / C;
    int c = (int)(gid - i * (long long)C);
    float di = dinv[i];
    float val = acc[gid] + H[gid] * (di * di) + bias[c];      // self-loop norm = dinv^2
    float inv = rsqrtf(var[c] + BN_EPS);
    float y = (val - mean[c]) * inv * gamma[c] + beta[c];
    out[gid] = fmaxf(y, 0.0f);
}

// ---------------- launch ----------------

static inline int blocks_for(long long total, int bs) {
    return (int)((total + bs - 1) / bs);
}

extern "C" void kernel_launch(void* const* d_in, const int* in_sizes, int n_in,
                              void* d_out, int out_size, void* d_ws, size_t ws_size,
                              hipStream_t stream) {
    const float*     x  = (const float*)d_in[0];
    const long long* ei = (const long long*)d_in[1];       // int64 edge_index [2, E]
    const float*     W1 = (const float*)d_in[2];
    const float*     b1 = (const float*)d_in[3];
    const float*     g1 = (const float*)d_in[4];
    const float*     be1= (const float*)d_in[5];
    const float*     m1 = (const float*)d_in[6];
    const float*     v1 = (const float*)d_in[7];
    const float*     W2 = (const float*)d_in[8];
    const float*     b2 = (const float*)d_in[9];
    const float*     g2 = (const float*)d_in[10];
    const float*     be2= (const float*)d_in[11];
    const float*     m2 = (const float*)d_in[12];
    const float*     v2 = (const float*)d_in[13];
    float* out = (float*)d_out;

    const int CIN = 128, CHID = 64, COUT = 40;
    const long long N = (long long)in_sizes[0] / CIN;      // 100000
    const long long E = (long long)in_sizes[1] / 2;        // 1600000
    const long long* src = ei;
    const long long* dst = ei + E;

    // Workspace layout (floats): deg[N] dinv[N] h1[N*64] acc1[N*64] h2[N*40] acc2[N*40]
    float* ws   = (float*)d_ws;
    float* deg  = ws;
    float* dinv = deg  + N;
    float* h1   = dinv + N;
    float* acc1 = h1   + N * CHID;     // after node_finish<64>, acc1 holds BN/ReLU output
    float* h2   = acc1 + N * CHID;
    float* acc2 = h2   + N * COUT;

    const int BS = 256;

    // degree (with self-loop) -> deg^{-1/2}
    fill_f32<<<blocks_for(N, BS), BS, 0, stream>>>(deg, N, 1.0f);
    deg_count<<<blocks_for(E, BS), BS, 0, stream>>>(dst, deg, E);
    make_dinv<<<blocks_for(N, BS), BS, 0, stream>>>(deg, dinv, N);

    // ---- layer 1 ----
    {   // h1 = x @ W1   (N x 128) @ (128 x 64)
        int colTiles = CHID / 16;                          // 4
        long long tiles = (N / 16) * colTiles;             // 25000
        gemm_wmma_f32<CHID><<<blocks_for(tiles, 8), BS, 0, stream>>>(
            x, W1, h1, (int)N, CIN, colTiles);
    }
    fill_f32<<<blocks_for(N * CHID, BS), BS, 0, stream>>>(acc1, N * CHID, 0.0f);
    edge_scatter<CHID><<<blocks_for(E * CHID, BS), BS, 0, stream>>>(
        h1, src, dst, dinv, acc1, E * CHID);
    node_finish<CHID><<<blocks_for(N * CHID, BS), BS, 0, stream>>>(
        h1, acc1, dinv, b1, g1, be1, m1, v1, acc1 /*in-place*/, N * CHID);

    // ---- layer 2 ----
    {   // h2 = acc1 @ W2   (N x 64) @ (64 x 40), col tiles padded to 48 w/ masking
        int colTiles = (COUT + 15) / 16;                   // 3
        long long tiles = (N / 16) * colTiles;             // 18750
        gemm_wmma_f32<COUT><<<blocks_for(tiles, 8), BS, 0, stream>>>(
            acc1, W2, h2, (int)N, CHID, colTiles);
    }
    fill_f32<<<blocks_for(N * COUT, BS), BS, 0, stream>>>(acc2, N * COUT, 0.0f);
    edge_scatter<COUT><<<blocks_for(E * COUT, BS), BS, 0, stream>>>(
        h2, src, dst, dinv, acc2, E * COUT);
    node_finish<COUT><<<blocks_for(N * COUT, BS), BS, 0, stream>>>(
        h2, acc2, dinv, b2, g2, be2, m2, v2, out, N * COUT);
}